// Net_24988119728916
// MI455X (gfx1250) — compile-verified
//
#include <hip/hip_runtime.h>
#include <hip/hip_bf16.h>
#include <math.h>

typedef float v2f __attribute__((ext_vector_type(2)));
typedef float v8f __attribute__((ext_vector_type(8)));

// ---------------------------------------------------------------------------
// GCN normalization kernels
// ---------------------------------------------------------------------------
__global__ __launch_bounds__(256) void deg_kernel(const int* __restrict__ col,
                                                  float* __restrict__ deg, int E) {
  int e = blockIdx.x * blockDim.x + threadIdx.x;
  if (e < E) atomicAdd(&deg[col[e]], 1.0f);
}

__global__ __launch_bounds__(256) void dinv_kernel(float* __restrict__ deg, int n) {
  int i = blockIdx.x * blockDim.x + threadIdx.x;
  if (i < n) {
    float d = deg[i];
    deg[i] = (d > 0.0f) ? rsqrtf(fmaxf(d, 1.0f)) : 0.0f;
  }
}

__global__ __launch_bounds__(256) void norm_kernel(const int* __restrict__ row,
                                                   const int* __restrict__ col,
                                                   const float* __restrict__ dinv,
                                                   float* __restrict__ nrm, int E) {
  int e = blockIdx.x * blockDim.x + threadIdx.x;
  if (e < E) nrm[e] = dinv[row[e]] * dinv[col[e]];
}

// ---------------------------------------------------------------------------
// Edge propagation: dst[col[e], :] += src[row[e], :] * norm[e]
// One thread per (edge, float4 chunk); chunk is the fast index so adjacent
// lanes cover one edge's feature row contiguously (b128 loads). The 19-26 MB
// feature buffers are L2-resident on MI455X (192 MB L2), so the
// global_atomic_add_f32 traffic resolves on-chip.
// ---------------------------------------------------------------------------
__global__ __launch_bounds__(256) void propagate_kernel(
    const float* __restrict__ src, float* __restrict__ dst,
    const int* __restrict__ row, const int* __restrict__ col,
    const float* __restrict__ nrm, int E, int nchunk, int ldf) {
  unsigned tid = blockIdx.x * blockDim.x + threadIdx.x;
  unsigned e = tid / (unsigned)nchunk;
  unsigned c = tid % (unsigned)nchunk;
  if (e >= (unsigned)E) return;
  int r = row[e];
  int cl = col[e];
  float nm = nrm[e];
  const float4 v = *(const float4*)(src + (size_t)r * ldf + (size_t)c * 4u);
  float* d = dst + (size_t)cl * ldf + (size_t)c * 4u;
  atomicAdd(d + 0, v.x * nm);
  atomicAdd(d + 1, v.y * nm);
  atomicAdd(d + 2, v.z * nm);
  atomicAdd(d + 3, v.w * nm);
}

// ---------------------------------------------------------------------------
// FP32 WMMA GEMM:  C[M, ldc] = A[M, lda] x W[kvalid, ldw(nvalid)]
// Compile-time NTILES/KDIM so the WMMA tile loop fully unrolls into fixed
// VGPR accumulators (no movrel), and the K-loop is a clean
// global_load_b64(A) + ds_load_b64(B) + v_wmma_f32_16x16x4_f32 stream.
// W is staged in LDS once per block, zero-padded over both invalid columns
// and invalid K rows, and packed K-pair-interleaved so each lane's B
// fragment {W[kk][n], W[kk+1][n]} is one ds_load_b64.
// Layouts per CDNA5 ISA:
//   A frag (16x4):  lanes 0-15 rows hold K={0,1}; lanes 16-31 hold K={2,3}
//   B frag (4x16):  vgpr0 = rows K0/K2 across lane halves, vgpr1 = K1/K3
//   C/D (16x16):    vgpr v: lanes 0-15 -> M=v, lanes 16-31 -> M=v+8
// MODE: 0 = plain store; 1 = relu(acc + addsrc + bias); 2 = MODE1 + mean acc.
// Requires M % 16 == 0 (true: N = 100000 = 16 * 6250).
// ---------------------------------------------------------------------------
template <int NTILES, int KDIM, int MODE>
__global__ __launch_bounds__(256) void gemm_wmma_kernel(
    const float* __restrict__ A, int lda, int kvalid,
    const float* __restrict__ W, int ldw, int nvalid,
    float* __restrict__ C, int ldc,
    const float* __restrict__ addsrc, const float* __restrict__ bias,
    float* __restrict__ meanDst, float meanScale, int M) {
  constexpr int NCOLS = NTILES * 16;
  __shared__ float wlds[KDIM * NCOLS];  // index: ((kk>>1)*NCOLS + n)*2 + (kk&1)

  for (int idx = threadIdx.x; idx < KDIM * NCOLS; idx += 256) {
    const int kk = idx / NCOLS;
    const int n = idx % NCOLS;
    const float v =
        (kk < kvalid && n < nvalid) ? W[(size_t)kk * ldw + n] : 0.0f;
    wlds[((kk >> 1) * NCOLS + n) * 2 + (kk & 1)] = v;
  }
  __syncthreads();

  const int lane = threadIdx.x & 31;
  const int wave = threadIdx.x >> 5;
  const int rowTile = blockIdx.x * 8 + wave;
  if (rowTile * 16 >= M) return;  // wave-uniform: EXEC all-ones around WMMA

  const int row0 = rowTile * 16;
  const int r = lane & 15;
  const int khalf = lane >> 4;

  v8f acc[NTILES] = {};
  const float* Arow = A + (size_t)(row0 + r) * lda;

#pragma unroll
  for (int k0 = 0; k0 < KDIM; k0 += 4) {
    const int kk = k0 + khalf * 2;          // even
    const v2f a = *(const v2f*)(Arow + kk); // global_load_b64
    const int kpair = kk >> 1;
#pragma unroll
    for (int t = 0; t < NTILES; ++t) {
      const v2f b =
          *(const v2f*)(&wlds[(kpair * NCOLS + t * 16 + r) * 2]); // ds_load_b64
      acc[t] = __builtin_amdgcn_wmma_f32_16x16x4_f32(
          false, a, false, b, (short)0, acc[t], false, false);
    }
  }

#pragma unroll
  for (int t = 0; t < NTILES; ++t) {
    const int cc = t * 16 + r;
    float bv = 0.0f;
    if constexpr (MODE >= 1) bv = (cc < nvalid) ? bias[cc] : 0.0f;
#pragma unroll
    for (int v = 0; v < 8; ++v) {
      const int rr = row0 + v + khalf * 8;
      const size_t idx = (size_t)rr * ldc + cc;
      float val = acc[t][v];
      if constexpr (MODE >= 1) {
        val += addsrc[idx] + bv;
        val = fmaxf(val, 0.0f);
      }
      C[idx] = val;
      if constexpr (MODE == 2) meanDst[idx] += val * meanScale;
    }
  }
}

// ---------------------------------------------------------------------------
// Row-wise log_softmax over ncls classes; one wave32 per node.
// ---------------------------------------------------------------------------
__global__ __launch_bounds__(256) void logsoftmax_kernel(
    const float* __restrict__ src, int lds, int ncls,
    float* __restrict__ out, int n) {
  const int w = (int)((blockIdx.x * blockDim.x + threadIdx.x) >> 5);
  const int lane = threadIdx.x & 31;
  if (w >= n) return;
  const float* rowp = src + (size_t)w * lds;
  float x0 = (lane < ncls) ? rowp[lane] : -INFINITY;
  float x1 = (lane + 32 < ncls) ? rowp[lane + 32] : -INFINITY;
  float m = fmaxf(x0, x1);
#pragma unroll
  for (int off = 16; off > 0; off >>= 1) m = fmaxf(m, __shfl_xor(m, off, 32));
  float s = 0.0f;
  if (lane < ncls) s += expf(x0 - m);
  if (lane + 32 < ncls) s += expf(x1 - m);
#pragma unroll
  for (int off = 16; off > 0; off >>= 1) s += __shfl_xor(s, off, 32);
  const float l = m + logf(s);
  if (lane < ncls) out[(size_t)w * ncls + lane] = x0 - l;
  if (lane + 32 < ncls) out[(size_t)w * ncls + lane + 32] = x1 - l;
}

// ---------------------------------------------------------------------------
extern "C" void kernel_launch(void* const* d_in, const int* in_sizes, int n_in,
                              void* d_out, int out_size, void* d_ws, size_t ws_size,
                              hipStream_t stream) {
  const float* x      = (const float*)d_in[0];
  const int*   ei     = (const int*)d_in[1];   // int32 (JAX demotes int64)
  const float* w1init = (const float*)d_in[2]; // [K,128,64]
  const float* w1     = (const float*)d_in[3]; // [1,K,64,64]
  const float* w1root = (const float*)d_in[4]; // [2,K,128,64]
  const float* b1     = (const float*)d_in[5]; // [2,K,1,64]
  const float* w2init = (const float*)d_in[6]; // [K,64,41]
  const float* w2     = (const float*)d_in[7]; // [1,K,41,41]
  const float* w2root = (const float*)d_in[8]; // [2,K,64,41]
  const float* b2     = (const float*)d_in[9]; // [2,K,1,41]

  const int Fin = 128, H = 64, Cc = 41, Cp = 48, K = 2;
  const int N = in_sizes[0] / Fin;
  const int E = in_sizes[1] / 2;

  // workspace carve-up (floats, 256B aligned)
  float* ws = (float*)d_ws;
  size_t off = 0;
  auto carve = [&](size_t nel) {
    float* p = ws + off;
    off += (nel + 63) & ~(size_t)63;
    return p;
  };
  float* deg   = carve((size_t)N);       // degree, then dinv in-place
  float* nrm   = carve((size_t)E);       // per-edge norm
  float* P     = carve((size_t)N * H);
  float* Q     = carve((size_t)N * H);   // propagate accumulator
  float* R     = carve((size_t)N * H);
  float* hmean = carve((size_t)N * H);   // layer-1 K-mean (= layer-2 input)
  float* M2    = carve((size_t)N * Cp);  // layer-2 K-mean

  const int* erow = ei;
  const int* ecol = ei + E;

  const dim3 blk(256);
  const unsigned eblocks = (unsigned)((E + 255) / 256);
  const unsigned nblocks = (unsigned)((N + 255) / 256);
  const unsigned gblocks = (unsigned)((((N + 15) / 16) + 7) / 8);

  // ---- GCN normalization -------------------------------------------------
  hipMemsetAsync(deg, 0, (size_t)N * sizeof(float), stream);
  deg_kernel<<<eblocks, blk, 0, stream>>>(ecol, deg, E);
  dinv_kernel<<<nblocks, blk, 0, stream>>>(deg, N);
  norm_kernel<<<eblocks, blk, 0, stream>>>(erow, ecol, deg, nrm, E);

  // ---- Layer 1: ARMA(Fin=128 -> H=64), T=2, mean over K ------------------
  hipMemsetAsync(hmean, 0, (size_t)N * H * sizeof(float), stream);
  {
    const unsigned pblocks = (unsigned)(((size_t)E * (H / 4) + 255) / 256);
    for (int k = 0; k < K; ++k) {
      // P = x @ w1_init[k]
      gemm_wmma_kernel<4, 128, 0><<<gblocks, blk, 0, stream>>>(
          x, Fin, Fin, w1init + (size_t)k * Fin * H, H, H,
          P, H, nullptr, nullptr, nullptr, 0.0f, N);
      // t = 0: propagate P; P = relu(Q + x@root0 + b0)
      hipMemsetAsync(Q, 0, (size_t)N * H * sizeof(float), stream);
      propagate_kernel<<<pblocks, blk, 0, stream>>>(P, Q, erow, ecol, nrm, E, H / 4, H);
      gemm_wmma_kernel<4, 128, 1><<<gblocks, blk, 0, stream>>>(
          x, Fin, Fin, w1root + ((size_t)0 * K + k) * Fin * H, H, H,
          P, H, Q, b1 + (0 * K + k) * H, nullptr, 0.0f, N);
      // t = 1: R = P @ w1[0,k]; propagate R; P = relu(Q + x@root1 + b1); hmean += P/K
      gemm_wmma_kernel<4, 64, 0><<<gblocks, blk, 0, stream>>>(
          P, H, H, w1 + (size_t)k * H * H, H, H,
          R, H, nullptr, nullptr, nullptr, 0.0f, N);
      hipMemsetAsync(Q, 0, (size_t)N * H * sizeof(float), stream);
      propagate_kernel<<<pblocks, blk, 0, stream>>>(R, Q, erow, ecol, nrm, E, H / 4, H);
      gemm_wmma_kernel<4, 128, 2><<<gblocks, blk, 0, stream>>>(
          x, Fin, Fin, w1root + ((size_t)1 * K + k) * Fin * H, H, H,
          P, H, Q, b1 + (1 * K + k) * H, hmean, 1.0f / K, N);
    }
  }
  // outer relu(mean of relu'd stacks) is the identity — skip it.

  // ---- Layer 2: ARMA(H=64 -> C=41 padded 48), T=2, mean over K -----------
  hipMemsetAsync(M2, 0, (size_t)N * Cp * sizeof(float), stream);
  {
    const unsigned pblocks = (unsigned)(((size_t)E * (Cp / 4) + 255) / 256);
    for (int k = 0; k < K; ++k) {
      // P = hmean @ w2_init[k]   (pad cols 41..47 stay zero)
      gemm_wmma_kernel<3, 64, 0><<<gblocks, blk, 0, stream>>>(
          hmean, H, H, w2init + (size_t)k * H * Cc, Cc, Cc,
          P, Cp, nullptr, nullptr, nullptr, 0.0f, N);
      // t = 0
      hipMemsetAsync(Q, 0, (size_t)N * Cp * sizeof(float), stream);
      propagate_kernel<<<pblocks, blk, 0, stream>>>(P, Q, erow, ecol, nrm, E, Cp / 4, Cp);
      gemm_wmma_kernel<3, 64, 1><<<gblocks, blk, 0, stream>>>(
          hmean, H, H, w2root + ((size_t)0 * K + k) * H * Cc, Cc, Cc,
          P, Cp, Q, b2 + (0 * K + k) * Cc, nullptr, 0.0f, N);
      // t = 1: R = P @ w2[0,k] (K-dim padded 41 -> 44; LDS fill zero-pads rows)
      gemm_wmma_kernel<3, 44, 0><<<gblocks, blk, 0, stream>>>(
          P, Cp, Cc, w2 + (size_t)k * Cc * Cc, Cc, Cc,
          R, Cp, nullptr, nullptr, nullptr, 0.0f, N);
      hipMemsetAsync(Q, 0, (size_t)N * Cp * sizeof(float), stream);
      propagate_kernel<<<pblocks, blk, 0, stream>>>(R, Q, erow, ecol, nrm, E, Cp / 4, Cp);
      gemm_wmma_kernel<3, 64, 2><<<gblocks, blk, 0, stream>>>(
          hmean, H, H, w2root + ((size_t)1 * K + k) * H * Cc, Cc, Cc,
          P, Cp, Q, b2 + (1 * K + k) * Cc, M2, 1.0f / K, N);
    }
  }

  // ---- log_softmax over classes ------------------------------------------
  const unsigned lsblocks = (unsigned)(((size_t)N * 32 + 255) / 256);
  logsoftmax_kernel<<<lsblocks, blk, 0, stream>>>(M2, Cp, Cc, (float*)d_out, N);
}